// DataWindowLoss_60198261621418
// MI455X (gfx1250) — compile-verified
//
#include <hip/hip_runtime.h>
#include <math.h>

// Charbonnier loss of 7x7 box-smoothed channel-summed diff, B=16 C=3 H=W=512.
// Fully fused, bandwidth-bound: one pass over x,y (100.7 MB compulsory ->
// ~4.3 us at 23.3 TB/s), all global traffic as B128 loads, horizontal 7-tap
// as banded GEMM on V_WMMA_F32_16X16X4_F32 (fp32 exact), vertical 7-tap +
// Charbonnier (v_sqrt_f32) + deterministic 2-stage reduction.

typedef __attribute__((ext_vector_type(2))) float v2f;
typedef __attribute__((ext_vector_type(8))) float v8f;

#define TILE  64
#define ZROWS 80   // 70 used (64 + 2*3 row halo), padded to 80 for WMMA row groups
#define ZCOLS 72   // col t = global col (c0-4+t): 18 aligned float4 groups exactly
#define HROWS 80
#define HCOLS 64

__global__ __launch_bounds__(256)
void charb_fused_kernel(const float* __restrict__ x,
                        const float* __restrict__ y,
                        float* __restrict__ partial)
{
    __shared__ __align__(16) float zt[ZROWS][ZCOLS]; // channel-summed diff (+halo)
    __shared__ __align__(16) float ht[HROWS][HCOLS]; // horizontally box-summed
    __shared__ float red[256];

    const int tid = threadIdx.x;
    const int b  = blockIdx.z;
    const int r0 = blockIdx.y * TILE;
    const int c0 = blockIdx.x * TILE;

    // ---- zero WMMA pad rows 70..79 (must be 0.0f, never garbage: 0*Inf=NaN)
    for (int i = tid; i < (ZROWS - 70) * ZCOLS; i += 256) (&zt[70][0])[i] = 0.0f;

    // ---- load z = sum_c (x - y), all B128: 70 rows x 18 float4 groups.
    // c0 is a multiple of 64 so every 4-col group is fully in or fully out
    // of the image -> zero-pad clamp is whole-group (matches conv zero pad).
    const long plane = 512L * 512L;
    const float* xb = x + (long)b * 3 * plane;
    const float* yb = y + (long)b * 3 * plane;
    for (int i = tid; i < 70 * 18; i += 256) {
        const int lr  = i / 18, g = i % 18;
        const int gr  = r0 - 3 + lr;
        const int gc4 = c0 - 4 + 4 * g;
        float4 z = make_float4(0.0f, 0.0f, 0.0f, 0.0f);
        if ((unsigned)gr < 512u && (unsigned)gc4 <= 508u) {
            const long o = (long)gr * 512 + gc4;
            const float4 a0 = *(const float4*)(xb + o);
            const float4 b0 = *(const float4*)(yb + o);
            const float4 a1 = *(const float4*)(xb + plane + o);
            const float4 b1 = *(const float4*)(yb + plane + o);
            const float4 a2 = *(const float4*)(xb + 2 * plane + o);
            const float4 b2 = *(const float4*)(yb + 2 * plane + o);
            z.x = (a0.x - b0.x) + (a1.x - b1.x) + (a2.x - b2.x);
            z.y = (a0.y - b0.y) + (a1.y - b1.y) + (a2.y - b2.y);
            z.z = (a0.z - b0.z) + (a1.z - b1.z) + (a2.z - b2.z);
            z.w = (a0.w - b0.w) + (a1.w - b1.w) + (a2.w - b2.w);
        }
        *(float4*)(&zt[lr][4 * g]) = z;
    }
    __syncthreads();

    // ---- horizontal 7-tap box sum: ht[r][c] = sum_{u=c+1..c+7} zt[r][u]
#if __has_builtin(__builtin_amdgcn_wmma_f32_16x16x4_f32)
    // Out(16x16) = In(16x24) x S(24x16), S[u,n] = 1 iff n+1 <= u <= n+7.
    // 20 tiles (5 row-groups x 4 col-groups) over 8 waves, 6 chained WMMAs each.
    {
        const int wave = tid >> 5;
        const int lane = tid & 31;
        const int lm   = lane & 15;   // M index for A, N index for B/D
        const int lh   = lane >> 4;   // lane half selects K/M-high per ISA layout
        for (int t = wave; t < 20; t += 8) {
            const int rg = t >> 2, cg = t & 3;
            v8f acc = {};
            #pragma unroll
            for (int blk = 0; blk < 6; ++blk) {
                const int u0 = 4 * blk + 2 * lh;        // K = 2h+v (ISA A layout)
                v2f a, bm;
                a.x = zt[rg * 16 + lm][cg * 16 + u0];
                a.y = zt[rg * 16 + lm][cg * 16 + u0 + 1];
                bm.x = (u0     >= lm + 1 && u0     <= lm + 7) ? 1.0f : 0.0f;
                bm.y = (u0 + 1 >= lm + 1 && u0 + 1 <= lm + 7) ? 1.0f : 0.0f;
                acc = __builtin_amdgcn_wmma_f32_16x16x4_f32(
                          false, a, false, bm, (short)0, acc, false, false);
            }
            // D layout: VGPR p, half h -> M = p + 8h, N = lm
            #pragma unroll
            for (int p = 0; p < 8; ++p) {
                ht[rg * 16 + p + 8 * lh][cg * 16 + lm] = acc[p];
            }
        }
    }
#else
    #pragma message("fp32 WMMA builtin unavailable: scalar horizontal filter")
    for (int i = tid; i < HROWS * HCOLS; i += 256) {
        const int r = i >> 6, c = i & 63;
        float s = 0.0f;
        if (r < 70) {
            #pragma unroll
            for (int j = 1; j < 8; ++j) s += zt[r][c + j];
        }
        ht[r][c] = s;
    }
#endif
    __syncthreads();

    // ---- vertical 7-tap + Charbonnier, float4 LDS reads, v_sqrt_f32
    float local = 0.0f;
    #pragma unroll
    for (int it = 0; it < 4; ++it) {
        const int idx = tid + 256 * it;
        const int r = idx >> 4;       // 0..63
        const int g = idx & 15;       // 4-col group
        float4 s = make_float4(0.0f, 0.0f, 0.0f, 0.0f);
        #pragma unroll
        for (int j = 0; j < 7; ++j) {
            const float4 h = *(const float4*)(&ht[r + j][4 * g]);
            s.x += h.x; s.y += h.y; s.z += h.z; s.w += h.w;
        }
        const float k = 1.0f / 49.0f;
        const float vx = s.x * k, vy = s.y * k, vz = s.z * k, vw = s.w * k;
        local += __builtin_amdgcn_sqrtf(vx * vx + 1e-6f)
               + __builtin_amdgcn_sqrtf(vy * vy + 1e-6f)
               + __builtin_amdgcn_sqrtf(vz * vz + 1e-6f)
               + __builtin_amdgcn_sqrtf(vw * vw + 1e-6f);
    }

    // ---- deterministic block reduction
    red[tid] = local;
    __syncthreads();
    for (int s = 128; s > 0; s >>= 1) {
        if (tid < s) red[tid] += red[tid + s];
        __syncthreads();
    }
    if (tid == 0) {
        partial[(blockIdx.z * gridDim.y + blockIdx.y) * gridDim.x + blockIdx.x] = red[0];
    }
}

__global__ __launch_bounds__(256)
void charb_final_reduce(const float* __restrict__ partial, float* __restrict__ out,
                        int n, float scale)
{
    __shared__ float red[256];
    float s = 0.0f;
    for (int i = threadIdx.x; i < n; i += 256) s += partial[i];
    red[threadIdx.x] = s;
    __syncthreads();
    for (int k = 128; k > 0; k >>= 1) {
        if (threadIdx.x < k) red[threadIdx.x] += red[threadIdx.x + k];
        __syncthreads();
    }
    if (threadIdx.x == 0) out[0] = red[0] * scale;
}

extern "C" void kernel_launch(void* const* d_in, const int* in_sizes, int n_in,
                              void* d_out, int out_size, void* d_ws, size_t ws_size,
                              hipStream_t stream) {
    (void)in_sizes; (void)n_in; (void)out_size; (void)ws_size;
    const float* x = (const float*)d_in[0];
    const float* y = (const float*)d_in[1];
    float* out     = (float*)d_out;
    float* partial = (float*)d_ws;          // 1024 floats = 4 KB scratch

    dim3 grid(512 / TILE, 512 / TILE, 16);  // 8 x 8 x 16 = 1024 blocks
    dim3 block(256);                        // 8 waves (wave32)
    charb_fused_kernel<<<grid, block, 0, stream>>>(x, y, partial);

    const int nblocks = (512 / TILE) * (512 / TILE) * 16;
    const float scale = 1.0f / (16.0f * 512.0f * 512.0f);
    charb_final_reduce<<<1, 256, 0, stream>>>(partial, out, nblocks, scale);
}